// MonotonicAlign_86294482911715
// MI455X (gfx1250) — compile-verified
//
#include <hip/hip_runtime.h>
#include <hip/hip_bf16.h>

typedef __attribute__((ext_vector_type(2))) float v2f;
typedef __attribute__((ext_vector_type(8))) float v8f;

#define MB 32      // batch
#define MD 192     // feature dim
#define MTX 256    // text length
#define MTY 1024   // mel length
#define NEGV (-1e9f)
// -0.5 * log(2*pi) * 192
#define LPCONST (-176.43619837529723f)

// ---------------------------------------------------------------------------
// Kernel 1: row norms  dst[b,t] = -0.5 * sum_d src[b,d,t]^2
// src layout [B, D, T]; consecutive threads -> consecutive t -> coalesced.
// ---------------------------------------------------------------------------
__global__ void ma_sumsq_kernel(const float* __restrict__ src,
                                float* __restrict__ dst, int T, int total) {
    int t = blockIdx.x * blockDim.x + threadIdx.x;
    if (t >= total) return;
    int b = t / T;
    int j = t - b * T;
    const float* p = src + (size_t)b * MD * T + j;
    float s = 0.0f;
#pragma unroll 4
    for (int d = 0; d < MD; ++d) {
        float v = p[(size_t)d * T];
        s += v * v;
    }
    dst[t] = -0.5f * s;
}

// ---------------------------------------------------------------------------
// Kernel 2: transposed log-prior via f32 WMMA.
//   lp[b, m(=y idx), n(=x idx)] = sum_k y[b,k,m]*x[b,k,n] + ysq[m] + xsq[n] + C
// One wave per 64(M) x 32(N) macro tile: acc[4][2] subtiles of 16x16,
// K stepped by 4 with V_WMMA_F32_16X16X4_F32.
// Fragment layouts (wave32):
//   A 16x4 : lane(lid,half): elem0 = A[M=lid, K=2*half], elem1 = K=2*half+1
//   B 4x16 : lane(lid,half): elem0 = B[K=2*half, N=lid], elem1 = K=2*half+1
//   C 16x16: vgpr r, lane(lid,half) = C[M = r + 8*half, N = lid]
// ---------------------------------------------------------------------------
__global__ void ma_prior_wmma_kernel(const float* __restrict__ x,
                                     const float* __restrict__ y,
                                     const float* __restrict__ xsq,
                                     const float* __restrict__ ysq,
                                     float* __restrict__ lp) {
    const int b  = blockIdx.z;
    const int x0 = blockIdx.x * 32;   // N origin (text axis)
    const int y0 = blockIdx.y * 64;   // M origin (mel axis)
    const int lane = threadIdx.x;
    const int lid  = lane & 15;
    const int half = lane >> 4;

    const float* X = x + (size_t)b * MD * MTX;
    const float* Y = y + (size_t)b * MD * MTY;

    v8f acc[4][2];
#pragma unroll
    for (int i = 0; i < 4; ++i)
#pragma unroll
        for (int j = 0; j < 2; ++j) acc[i][j] = (v8f){0,0,0,0,0,0,0,0};

#pragma unroll 2
    for (int k0 = 0; k0 < MD; k0 += 4) {
        const int kk = k0 + 2 * half;
        v2f a[4], bb[2];
#pragma unroll
        for (int i = 0; i < 4; ++i) {
            const int m = y0 + 16 * i + lid;
            a[i].x = Y[(size_t)kk * MTY + m];
            a[i].y = Y[(size_t)(kk + 1) * MTY + m];
        }
#pragma unroll
        for (int j = 0; j < 2; ++j) {
            const int n = x0 + 16 * j + lid;
            bb[j].x = X[(size_t)kk * MTX + n];
            bb[j].y = X[(size_t)(kk + 1) * MTX + n];
        }
#pragma unroll
        for (int i = 0; i < 4; ++i)
#pragma unroll
            for (int j = 0; j < 2; ++j)
                acc[i][j] = __builtin_amdgcn_wmma_f32_16x16x4_f32(
                    false, a[i], false, bb[j], (short)0, acc[i][j],
                    false, false);
    }

    // epilogue: add norms + const, store [B, TY, TX] (coalesced 64B runs)
    const float* xsqb = xsq + b * MTX;
    const float* ysqb = ysq + b * MTY;
    float* lpb = lp + (size_t)b * MTY * MTX;
#pragma unroll
    for (int i = 0; i < 4; ++i) {
#pragma unroll
        for (int r = 0; r < 8; ++r) {
            const int m = y0 + 16 * i + 8 * half + r;
            const float ys = ysqb[m];
#pragma unroll
            for (int j = 0; j < 2; ++j) {
                const int n = x0 + 16 * j + lid;
                lpb[(size_t)m * MTX + n] = acc[i][j][r] + ys + xsqb[n] + LPCONST;
            }
        }
    }
}

// ---------------------------------------------------------------------------
// Kernel 3: forward Viterbi DP + backtrack. One wave (32 lanes) per batch.
// Lane L owns x = 8L..8L+7 in registers. Per column y:
//   v_new[x] = lp[y,x] + max( (x==y ? NEG : v[x]), v[x-1] )
// decision bit d[y][x] = (x==y) | (v[x-1] > v[x])  packed 8 bits/lane into LDS
// (1024 * 32 B = 32 KB). Lane 0 then backtracks via LDS bits, emitting
// path[b][y] = active ? index : -1.
// ---------------------------------------------------------------------------
__global__ void ma_viterbi_kernel(const float* __restrict__ lp,
                                  const int* __restrict__ x_lens,
                                  const int* __restrict__ y_lens,
                                  int* __restrict__ path) {
    __shared__ unsigned char dbits[MTY * 32];  // 32 KB decision bits
    const int b = blockIdx.x;
    const int lane = threadIdx.x;
    const float* lpb = lp + (size_t)b * MTY * MTX;

    // init column y = 0 : v[x] = (x==0) ? lp[0,x] : NEG
    float v[8];
    {
        const float4 c0 = *(const float4*)(lpb + 8 * lane);
        v[0] = (lane == 0) ? c0.x : NEGV;
#pragma unroll
        for (int e = 1; e < 8; ++e) v[e] = NEGV;
    }

    for (int yv = 1; yv < MTY; ++yv) {
        const float4 c0 = *(const float4*)(lpb + (size_t)yv * MTX + 8 * lane);
        const float4 c1 = *(const float4*)(lpb + (size_t)yv * MTX + 8 * lane + 4);
        float lc[8] = {c0.x, c0.y, c0.z, c0.w, c1.x, c1.y, c1.z, c1.w};

        // shifted previous values (v[x-1])
        float pv[8];
        float up = __shfl_up(v[7], 1, 32);
        pv[0] = (lane == 0) ? NEGV : up;
#pragma unroll
        for (int e = 1; e < 8; ++e) pv[e] = v[e - 1];

        unsigned bits = 0u;
#pragma unroll
        for (int e = 0; e < 8; ++e) {
            const int xv = 8 * lane + e;
            const bool diag = (xv == yv);
            if (diag || (pv[e] > v[e])) bits |= (1u << e);
            const float vc = diag ? NEGV : v[e];
            v[e] = lc[e] + fmaxf(vc, pv[e]);
        }
        dbits[yv * 32 + lane] = (unsigned char)bits;
    }
    __syncthreads();

    if (lane == 0) {
        const int yl = y_lens[b];
        int idx = x_lens[b] - 1;
        int* pth = path + b * MTY;
        for (int yv = MTY - 1; yv >= 0; --yv) {
            const bool active = (yv < yl);
            pth[yv] = active ? idx : -1;
            if (active && (yv > 0) && (idx != 0)) {
                const unsigned byv = dbits[yv * 32 + (idx >> 3)];
                idx -= (int)((byv >> (idx & 7)) & 1u);
            }
        }
    }
}

// ---------------------------------------------------------------------------
// Kernel 4: expand compact path to one-hot attn [B, TX, TY] (f32 out).
// ---------------------------------------------------------------------------
__global__ void ma_emit_kernel(const int* __restrict__ path,
                               float* __restrict__ out) {
    const size_t idx = (size_t)blockIdx.x * blockDim.x + threadIdx.x;
    if (idx >= (size_t)MB * MTX * MTY) return;
    const int yv = (int)(idx & (MTY - 1));
    const size_t bx = idx >> 10;          // b*TX + x
    const int xv = (int)(bx & (MTX - 1));
    const int b  = (int)(bx >> 8);
    out[idx] = (path[b * MTY + yv] == xv) ? 1.0f : 0.0f;
}

// ---------------------------------------------------------------------------
extern "C" void kernel_launch(void* const* d_in, const int* in_sizes, int n_in,
                              void* d_out, int out_size, void* d_ws, size_t ws_size,
                              hipStream_t stream) {
    const float* x      = (const float*)d_in[0];   // [B, D, TX]
    const int*   x_lens = (const int*)d_in[1];     // [B]
    const float* y      = (const float*)d_in[2];   // [B, D, TY]
    const int*   y_lens = (const int*)d_in[3];     // [B]
    float* out = (float*)d_out;                    // [B, TX, TY]

    // workspace layout
    float* lp   = (float*)d_ws;                        // B*TY*TX f32 = 32 MB
    float* ysq  = lp + (size_t)MB * MTY * MTX;         // B*TY
    float* xsq  = ysq + (size_t)MB * MTY;              // B*TX
    int*   path = (int*)(xsq + (size_t)MB * MTX);      // B*TY ints

    // 1) row norms
    {
        int totx = MB * MTX;
        ma_sumsq_kernel<<<(totx + 255) / 256, 256, 0, stream>>>(x, xsq, MTX, totx);
        int toty = MB * MTY;
        ma_sumsq_kernel<<<(toty + 255) / 256, 256, 0, stream>>>(y, ysq, MTY, toty);
    }
    // 2) transposed log-prior via WMMA: grid = (TX/32, TY/64, B), 1 wave/block
    {
        dim3 grid(MTX / 32, MTY / 64, MB);
        ma_prior_wmma_kernel<<<grid, 32, 0, stream>>>(x, y, xsq, ysq, lp);
    }
    // 3) Viterbi DP + backtrack: one wave per batch
    ma_viterbi_kernel<<<MB, 32, 0, stream>>>(lp, x_lens, y_lens, path);

    // 4) expand to one-hot output
    {
        size_t tot = (size_t)MB * MTX * MTY;
        ma_emit_kernel<<<(unsigned)((tot + 255) / 256), 256, 0, stream>>>(path, out);
    }
}